// NeighSuperpixelAgg_81071802680046
// MI455X (gfx1250) — compile-verified
//
#include <hip/hip_runtime.h>
#include <hip/hip_bf16.h>

typedef __attribute__((ext_vector_type(2))) float v2f;
typedef __attribute__((ext_vector_type(8))) float v8f;
typedef __attribute__((ext_vector_type(4))) unsigned int v4u;
typedef __attribute__((ext_vector_type(8))) int v8i;
typedef __attribute__((ext_vector_type(4))) int v4i;

// ---------------------------------------------------------------------------
// TDM: DMA a contiguous 128x128 fp32 tile from global memory into LDS.
// Descriptor per CDNA5 ISA ch.8 (Tensor DMA Descriptor D#):
//  group0: count=1 | lds_addr | global_addr[56:0] | type=2
//  group1: data_size=4B, tensor_dim0=128, tensor_dim1=128, tile_dim0=128,
//          tile_dim1=128, tensor_dim0_stride=128  (2D -> groups 2/3 zero)
// Issued by a single wave; completion via s_wait_tensorcnt 0.
// This toolchain exposes the 6-arg builtin: (v4u, v8i, v4i, v4i, v8i, i32).
// ---------------------------------------------------------------------------
__device__ __forceinline__ void tdm_load_tile_128x128_f32(
    const float* __restrict__ gsrc, unsigned lds_byte_off)
{
  const unsigned long long ga = (unsigned long long)(uintptr_t)gsrc;

  v4u g0;
  g0[0] = 1u;                                       // count=1 (valid user D#)
  g0[1] = lds_byte_off;                             // lds_addr (bytes)
  g0[2] = (unsigned)(ga & 0xFFFFFFFFu);             // global_addr[31:0]
  g0[3] = (unsigned)((ga >> 32) & 0x01FFFFFFu)      // global_addr[56:32]
        | 0x80000000u;                              // type=2 ("image") [127:126]

  v8i g1;
  g1[0] = 0x00020000;          // workgroup_mask=0, data_size=2 (4 bytes)
  g1[1] = (int)(128u << 16);   // [47:32] atomic_barrier_addr=0, [63:48] tensor_dim0.lo=128
  g1[2] = (int)(128u << 16);   // [79:64] tensor_dim0.hi=0, [95:80] tensor_dim1.lo=128
  g1[3] = (int)(128u << 16);   // [111:96] tensor_dim1.hi=0, [127:112] tile_dim0=128
  g1[4] = 128;                 // [143:128] tile_dim1=128, [159:144] tile_dim2=0
  g1[5] = 128;                 // [191:160] tensor_dim0_stride.lo=128
  g1[6] = 0;                   // stride.hi=0, tensor_dim1_stride unused (2D)
  g1[7] = 0;

  const v4i gz4 = {0, 0, 0, 0};            // 2D tensor: groups 2/3 not needed
  const v8i gz8 = {0, 0, 0, 0, 0, 0, 0, 0};

  __builtin_amdgcn_tensor_load_to_lds(g0, g1, gz4, gz4, gz8, 0 /*cpol*/);
}

// ---------------------------------------------------------------------------
// GEMM + bias:  out[M,128] = A[M,128] @ W[128,128] + bias[128]
// One block = 128 (M) x 128 (N) tile. 256 threads = 8 waves (wave32).
// Wave w computes rows [16w, 16w+16), all 128 columns, as 8 WMMA 16x16 tiles.
// W is DMA'd into LDS by the Tensor Data Mover (wave 0) while every wave
// preloads its A strip into registers (32 x v2f fragments); fp32 WMMA
// (V_WMMA_F32_16X16X4_F32) -- the problem is HBM-bound, keep full precision.
// ---------------------------------------------------------------------------
__global__ __launch_bounds__(256) void gemm_bias_wmma_f32(
    const float* __restrict__ A, const float* __restrict__ W,
    const float* __restrict__ bias, float* __restrict__ out)
{
  __shared__ float Bs[128 * 128];           // 64 KB

  const int tid  = threadIdx.x;             // 0..255
  const int wave = tid >> 5;                // 0..7
  const int lane = tid & 31;
  const int lo   = lane & 15;               // lane 0-15 index
  const int hi   = lane >> 4;               // 0 or 1 (lane half)
  const long m0  = (long)blockIdx.x * 128;  // block's first output row

  // ---- issue TDM for W -> LDS from wave 0 (flat addr low 32 bits = LDS off)
  if (tid < 32) {
    const unsigned lds_off = (unsigned)(uintptr_t)&Bs[0];
    tdm_load_tile_128x128_f32(W, lds_off);
  }

  // ---- overlap: preload this wave's A fragments into registers ------------
  // A fragment layout (16x4 f32, MxK): lane l, VGPR v holds A[l&15][2*(l>>4)+v]
  const int    r0   = wave * 16;
  const float* Arow = A + (m0 + r0 + lo) * 128 + 2 * hi;
  v2f areg[32];
#pragma unroll
  for (int kk = 0; kk < 32; ++kk)
    areg[kk] = *(const v2f*)(Arow + 4 * kk);   // 8B aligned (even offset)

  // ---- wait for the tensor DMA, then workgroup barrier --------------------
  if (tid < 32) __builtin_amdgcn_s_wait_tensorcnt(0);
  __syncthreads();

  // ---- compute 8 16x16 output tiles ---------------------------------------
#pragma unroll
  for (int n = 0; n < 8; ++n) {
    // C/D layout: VGPR r, lane l -> D[r + 8*(l>>4)][l&15]; bias is per-column
    const float bn = bias[n * 16 + lo];
    v8f acc;
#pragma unroll
    for (int r = 0; r < 8; ++r) acc[r] = bn;

    // B fragment (4x16 f32, KxN): lane l, VGPR v holds B[k + 2*(l>>4) + v][n*16 + (l&15)]
    const float* Bcol = Bs + (2 * hi) * 128 + n * 16 + lo;
#pragma unroll
    for (int k = 0; k < 128; k += 4) {
      v2f bfrag;
      bfrag.x = Bcol[k * 128];
      bfrag.y = Bcol[k * 128 + 128];
      acc = __builtin_amdgcn_wmma_f32_16x16x4_f32(
          /*neg_a=*/false, areg[k >> 2],
          /*neg_b=*/false, bfrag,
          /*c_mod=*/(short)0, acc,
          /*reuse_a=*/false, /*reuse_b=*/false);
    }

    float* orow = out + (m0 + r0 + 8 * hi) * 128 + n * 16 + lo;
#pragma unroll
    for (int r = 0; r < 8; ++r) orow[(long)r * 128] = acc[r];
  }
}

// ---------------------------------------------------------------------------
// Superpixel-gated 7x7 neighborhood aggregation (NATTEN-style clamped window)
//   out[b,i,j,c] = sum_{ki,kj} attn[b,h,i,j,ki*7+kj] * gate(ni,nj) * v[b,ni,nj,c]
// One block per pixel, 128 threads = 128 channels (head = c>>5).
// attn weights (4x49) and gates (49) staged in LDS once per pixel.
// v (67 MB) stays resident in the 192 MB L2 across the 49x reuse.
// ---------------------------------------------------------------------------
__global__ __launch_bounds__(128) void nsa_agg_kernel(
    const float* __restrict__ v, const float* __restrict__ attn,
    const int* __restrict__ sp, float* __restrict__ out)
{
  __shared__ float sa[4 * 49];
  __shared__ float sg[49];

  const int  tid = threadIdx.x;            // channel 0..127
  const long pix = blockIdx.x;             // b*H*W + i*W + j
  const int  j   = (int)(pix & 127);
  const int  i   = (int)((pix >> 7) & 127);
  const int  b   = (int)(pix >> 14);

  const int si = min(max(i - 3, 0), 121);  // clamp window start (H-k = 121)
  const int sj = min(max(j - 3, 0), 121);

  const int* spb    = sp + (long)b * (128 * 128);
  const int  center = spb[i * 128 + j];

  // attn flat index: ((((b*4 + h)*128 + i)*128 + j)*49 + t)
  const float* abase = attn + (((long)(b * 4) * 128 + i) * 128 + j) * 49;
  const long   hstride = (long)128 * 128 * 49;
  for (int t = tid; t < 4 * 49; t += 128) {
    const int h = t / 49;
    const int w = t - h * 49;
    sa[t] = abase[(long)h * hstride + w];
  }
  if (tid < 49) {
    const int ki = tid / 7;
    const int kj = tid - ki * 7;
    sg[tid] = (spb[(si + ki) * 128 + (sj + kj)] == center) ? 1.0f : 0.0f;
  }
  __syncthreads();

  const int    head = tid >> 5;
  const float* vb   = v + (long)b * (128 * 128 * 128);

  float acc = 0.0f;
#pragma unroll
  for (int ki = 0; ki < 7; ++ki) {
    const float* vrow = vb + ((long)(si + ki) * 128 + sj) * 128 + tid;
#pragma unroll
    for (int kj = 0; kj < 7; ++kj) {
      const float w = sa[head * 49 + ki * 7 + kj] * sg[ki * 7 + kj];
      acc = fmaf(w, vrow[kj * 128], acc);
    }
  }
  out[pix * 128 + tid] = acc;
}

// ---------------------------------------------------------------------------
extern "C" void kernel_launch(void* const* d_in, const int* in_sizes, int n_in,
                              void* d_out, int out_size, void* d_ws, size_t ws_size,
                              hipStream_t stream) {
  const float* x    = (const float*)d_in[0];  // [8,128,128,128]
  const float* attn = (const float*)d_in[1];  // [8,4,128,128,49]
  const int*   sp   = (const int*)  d_in[2];  // [8,128,128]
  const float* Wv   = (const float*)d_in[3];  // [128,128]
  const float* bv   = (const float*)d_in[4];  // [128]
  const float* Wp   = (const float*)d_in[5];  // [128,128]
  const float* bp   = (const float*)d_in[6];  // [128]
  float*       out  = (float*)d_out;          // [8,128,128,128]

  const long M = 8L * 128 * 128;              // 131072 pixels
  float* vws   = (float*)d_ws;                // [M,128] v = x@Wv+bv
  float* aggws = vws + M * 128;               // [M,128] aggregated

  // 1) v = x @ Wv + bv
  gemm_bias_wmma_f32<<<(unsigned)(M / 128), 256, 0, stream>>>(x, Wv, bv, vws);
  // 2) superpixel-gated neighborhood aggregation
  nsa_agg_kernel<<<(unsigned)M, 128, 0, stream>>>(vws, attn, sp, aggws);
  // 3) y = agg @ Wp + bp
  gemm_bias_wmma_f32<<<(unsigned)(M / 128), 256, 0, stream>>>(aggws, Wp, bp, out);
}